// S4DKernel_17351667876169
// MI455X (gfx1250) — compile-verified
//
#include <hip/hip_runtime.h>

typedef __attribute__((ext_vector_type(2))) float v2f;
typedef __attribute__((ext_vector_type(8))) float v8f;

#define H_DIM     512
#define NH        32          // complex modes per h
#define SEQ_L     8192
#define B_TILE    16          // N dim of WMMA tile (l % 16)
#define A_PER_H   (SEQ_L / B_TILE)       // 512 "a" rows per h
#define M_TILES   (A_PER_H / 16)         // 32 M-tiles of 16 rows
#define WAVES     8
#define TPW       (M_TILES / WAVES)      // 4 tiles per wave

// sin/cos with cheap exact-enough range reduction (error ~5e-7 rad, far below
// the ~8e-3 rad f32 phase quantization inherent to the reference itself).
__device__ __forceinline__ void sincos_red(float th, float* s, float* c) {
    const float INV2PI = 0.15915494309189535f;
    const float C1     = 6.2831854820251465f;   // fl32(2*pi)
    const float C2     = -1.7484556000744953e-7f; // 2*pi - C1
    float k = rintf(th * INV2PI);
    float r = fmaf(-k, C1, th);
    r = fmaf(-k, C2, r);
    __sincosf(r, s, c);                          // v_sin_f32 / v_cos_f32
}

// K[h, 16a+b] = sum_k A[a,k] * B[k,b]   (per-h GEMM: 512x64 @ 64x16)
//   A[a, k<32 ] =  P.re[n=k,   a],  A[a, k>=32] = -P.im[n=k-32, a]
//   B[k<32 , b] =  Q.re[n=k,   b],  B[k>=32, b] =  Q.im[n=k-32, b]
//   P[n,a] = 2*C'[n]*exp(dtA[n]*16a),  Q[n,b] = exp(dtA[n]*b)
__global__ __launch_bounds__(256) void s4d_wmma_kernel(
    const float* __restrict__ Cin,        // (1, H, NH, 2)
    const float* __restrict__ log_dt,     // (H)
    const float* __restrict__ log_A_real, // (H, NH)
    const float* __restrict__ A_imag,     // (H, NH)
    float* __restrict__ out)              // (1, H, SEQ_L)
{
    __shared__ float s_dre[NH], s_dim[NH];     // dtA per mode
    __shared__ float s_cre[NH], s_cim[NH];     // 2*C' per mode
    __shared__ float s_B[2 * NH][B_TILE];      // B matrix [k][b]

    const int h   = blockIdx.x;
    const int tid = threadIdx.x;

    // ---- Phase 1: per-mode coefficients ----
    if (tid < NH) {
        const int n = tid;
        const float dt   = __expf(log_dt[h]);
        const float Are  = -__expf(log_A_real[h * NH + n]);
        const float Aim  = A_imag[h * NH + n];
        const float dre  = Are * dt;
        const float dimm = Aim * dt;
        // f = (exp(dtA) - 1) / A
        float sn, cs;
        const float e = __expf(dre);
        sincos_red(dimm, &sn, &cs);
        const float numre = e * cs - 1.0f;
        const float numim = e * sn;
        const float inv   = 1.0f / (Are * Are + Aim * Aim);
        const float fre   = (numre * Are + numim * Aim) * inv;
        const float fim   = (numim * Are - numre * Aim) * inv;
        const float Cre   = Cin[(h * NH + n) * 2 + 0];
        const float Cim   = Cin[(h * NH + n) * 2 + 1];
        s_dre[n] = dre;
        s_dim[n] = dimm;
        s_cre[n] = 2.0f * (Cre * fre - Cim * fim);   // 2*C'
        s_cim[n] = 2.0f * (Cre * fim + Cim * fre);
    }
    __syncthreads();

    // ---- Phase 2: B table  Q[n,b] = exp(dtA[n]*b), 512 complex entries ----
    {
        int idx = tid;
#pragma unroll
        for (int r = 0; r < 2; ++r, idx += 256) {
            const int   n  = idx >> 4;
            const int   b  = idx & 15;
            const float fb = (float)b;
            float sn, cs;
            const float e = __expf(s_dre[n] * fb);
            sincos_red(s_dim[n] * fb, &sn, &cs);
            s_B[n][b]      = e * cs;   // Q.re
            s_B[NH + n][b] = e * sn;   // Q.im
        }
    }
    __syncthreads();

    // ---- Phase 3: WMMA GEMM ----
    const int lane = tid & 31;
    const int wave = tid >> 5;
    const int hi   = lane >> 4;    // lane half: selects K pair {0,1} vs {2,3}
    const int mrow = lane & 15;    // A-matrix row within tile
    const int bcol = lane & 15;    // B/D column

    // Tile-invariant per-lane data: hoist out of the tile loop into registers.
    // Lane j-th K-slot maps to mode n(j) = 4*(j/2) + 2*hi + (j&1).
    float dreg[16], dimg[16], creg[16], cimg[16], breg[16], bimg[16];
#pragma unroll
    for (int j = 0; j < 16; ++j) {
        const int n = 4 * (j >> 1) + 2 * hi + (j & 1);
        dreg[j] = s_dre[n];
        dimg[j] = s_dim[n];
        creg[j] = s_cre[n];
        cimg[j] = s_cim[n];
        breg[j] = s_B[n][bcol];
        bimg[j] = s_B[NH + n][bcol];
    }

    for (int t = 0; t < TPW; ++t) {
        const int   a0 = (wave * TPW + t) * 16;
        const float fa = 16.0f * (float)(a0 + mrow);   // l-offset of this row

        v8f acc = {0.f, 0.f, 0.f, 0.f, 0.f, 0.f, 0.f, 0.f};
        float pimneg[16];   // -P.im, cached for the second K-half

        // K-steps 0..7: real half (A = P.re, B = Q.re)
#pragma unroll
        for (int kk = 0; kk < 8; ++kk) {
            const int j0 = 2 * kk, j1 = j0 + 1;
            float s0, c0, s1, c1;
            const float e0 = __expf(dreg[j0] * fa);
            sincos_red(dimg[j0] * fa, &s0, &c0);
            const float er0 = e0 * c0, ei0 = e0 * s0;
            const float pr0 = fmaf(creg[j0], er0, -(cimg[j0] * ei0));
            pimneg[j0]      = fmaf(-creg[j0], ei0, -(cimg[j0] * er0));

            const float e1 = __expf(dreg[j1] * fa);
            sincos_red(dimg[j1] * fa, &s1, &c1);
            const float er1 = e1 * c1, ei1 = e1 * s1;
            const float pr1 = fmaf(creg[j1], er1, -(cimg[j1] * ei1));
            pimneg[j1]      = fmaf(-creg[j1], ei1, -(cimg[j1] * er1));

            v2f afrag; afrag.x = pr0;      afrag.y = pr1;
            v2f bfrag; bfrag.x = breg[j0]; bfrag.y = breg[j1];
            acc = __builtin_amdgcn_wmma_f32_16x16x4_f32(
                false, afrag, false, bfrag, (short)0, acc, false, false);
        }

        // K-steps 8..15: imaginary half (A = -P.im, B = Q.im)
#pragma unroll
        for (int kk = 0; kk < 8; ++kk) {
            const int j0 = 2 * kk, j1 = j0 + 1;
            v2f afrag; afrag.x = pimneg[j0]; afrag.y = pimneg[j1];
            v2f bfrag; bfrag.x = bimg[j0];   bfrag.y = bimg[j1];
            acc = __builtin_amdgcn_wmma_f32_16x16x4_f32(
                false, afrag, false, bfrag, (short)0, acc, false, false);
        }

        // ---- Store D: lane holds rows M = v + 8*hi, col = bcol ----
        float* op = out + ((size_t)h << 13);
        const int mbase = a0 + (hi << 3);
#pragma unroll
        for (int v = 0; v < 8; ++v) {
            const int l = ((mbase + v) << 4) + bcol;
            op[l] = acc[v];
        }
    }
}

extern "C" void kernel_launch(void* const* d_in, const int* in_sizes, int n_in,
                              void* d_out, int out_size, void* d_ws, size_t ws_size,
                              hipStream_t stream) {
    (void)in_sizes; (void)n_in; (void)out_size; (void)d_ws; (void)ws_size;
    const float* C           = (const float*)d_in[0];
    const float* log_dt      = (const float*)d_in[1];
    const float* log_A_real  = (const float*)d_in[2];
    const float* A_imag      = (const float*)d_in[3];
    // d_in[4] is L == 8192 (compile-time constant here)
    float* out = (float*)d_out;

    s4d_wmma_kernel<<<H_DIM, 256, 0, stream>>>(C, log_dt, log_A_real, A_imag, out);
}